// ModelNew_25056839205286
// MI455X (gfx1250) — compile-verified
//
#include <hip/hip_runtime.h>
#include <hip/hip_bf16.h>
#include <math.h>

typedef float v2f __attribute__((ext_vector_type(2)));
typedef float v8f __attribute__((ext_vector_type(8)));

#define B_    32
#define NQ_   1024
#define NK_   49
#define D_    512
#define H_    8
#define DH_   64
#define QBLK  64
#define NWAVE 4

__device__ __forceinline__ v8f wmma4(v2f a, v2f b, v8f c) {
  // D = A(16x4 f32) * B(4x16 f32) + C(16x16 f32)
  return __builtin_amdgcn_wmma_f32_16x16x4_f32(false, a, false, b, (short)0, c,
                                               false, false);
}

__global__ __launch_bounds__(128) void attn_fwd_f32wmma(
    const float* __restrict__ Q, const float* __restrict__ K,
    const float* __restrict__ V, float* __restrict__ O) {
  __shared__ __align__(16) float Qlds[QBLK * DH_];      // [q][d], pre-scaled
  __shared__ __align__(16) float Klds[64 * DH_];        // [k][d], zero-padded
  __shared__ __align__(16) float Vlds[DH_ * 64];        // [d][k], zero-padded
  __shared__ __align__(16) float Plds[NWAVE][16 * 64];  // per-wave [q][k]

  const int qblk = blockIdx.x;
  const int h    = blockIdx.y;
  const int b    = blockIdx.z;
  const int tid  = threadIdx.x;
  const int qbase = qblk * QBLK;
  const float scale = 0.125f;  // 1/sqrt(Dh), Dh=64

  const float* Qg = Q + (b * NQ_ + qbase) * D_ + h * DH_;
  const float* Kg = K + (b * NK_) * D_ + h * DH_;
  const float* Vg = V + (b * NK_) * D_ + h * DH_;

  // ---- stage Q (scaled), coalesced float4 ----
  for (int i = tid; i < QBLK * DH_ / 4; i += 128) {
    int r = i >> 4, dv = (i & 15) << 2;
    float4 x = *(const float4*)(Qg + r * D_ + dv);
    x.x *= scale; x.y *= scale; x.z *= scale; x.w *= scale;
    *(float4*)&Qlds[r * DH_ + dv] = x;
  }
  // ---- stage K, zero-pad rows 49..63 ----
  for (int i = tid; i < 64 * DH_ / 4; i += 128) {
    int r = i >> 4, dv = (i & 15) << 2;
    float4 x = make_float4(0.f, 0.f, 0.f, 0.f);
    if (r < NK_) x = *(const float4*)(Kg + r * D_ + dv);
    *(float4*)&Klds[r * DH_ + dv] = x;
  }
  // ---- stage V transposed [d][k], zero-pad ----
  for (int i = tid; i < 64 * DH_ / 4; i += 128) {
    int r = i >> 4, dv = (i & 15) << 2;
    float4 x = make_float4(0.f, 0.f, 0.f, 0.f);
    if (r < NK_) x = *(const float4*)(Vg + r * D_ + dv);
    Vlds[(dv + 0) * 64 + r] = x.x;
    Vlds[(dv + 1) * 64 + r] = x.y;
    Vlds[(dv + 2) * 64 + r] = x.z;
    Vlds[(dv + 3) * 64 + r] = x.w;
  }
  __syncthreads();

  const int wv   = tid >> 5;
  const int lane = tid & 31;
  const int nn   = lane & 15;  // N-column / A-row index
  const int g    = lane >> 4;  // lane half-group

  // ---- A-fragments of Q: lane (m=nn,g) holds Q[m][4s+2g+{0,1}] ----
  v2f aq[16];
#pragma unroll
  for (int s = 0; s < 16; ++s)
    aq[s] = *(const v2f*)&Qlds[(wv * 16 + nn) * DH_ + 4 * s + 2 * g];

  // ---- S = (Q*scale) * K^T : 4 key tiles x 16 K-steps of f32 WMMA ----
  v8f sc[4];
#pragma unroll
  for (int t = 0; t < 4; ++t) {
    v8f acc = {0.f, 0.f, 0.f, 0.f, 0.f, 0.f, 0.f, 0.f};
#pragma unroll
    for (int s = 0; s < 16; ++s) {
      v2f bk = *(const v2f*)&Klds[(16 * t + nn) * DH_ + 4 * s + 2 * g];
      acc = wmma4(aq[s], bk, acc);
    }
    sc[t] = acc;
  }

  // ---- masked row softmax directly on the C/D layout ----
  // row m = r + 8*g lives in lane half-group g; columns are (tile t, lane nn)
#pragma unroll
  for (int r = 0; r < 8; ++r) {
    float sv[4];
    float mx = -1e30f;
#pragma unroll
    for (int t = 0; t < 4; ++t) {
      float x = sc[t][r];
      if (16 * t + nn >= NK_) x = -1e30f;  // padded keys
      sv[t] = x;
      mx = fmaxf(mx, x);
    }
#pragma unroll
    for (int off = 1; off < 16; off <<= 1)
      mx = fmaxf(mx, __shfl_xor(mx, off, 32));
    float sum = 0.f;
#pragma unroll
    for (int t = 0; t < 4; ++t) {
      float e = __expf(sv[t] - mx);
      sv[t] = e;
      sum += e;
    }
#pragma unroll
    for (int off = 1; off < 16; off <<= 1)
      sum += __shfl_xor(sum, off, 32);
    float inv = 1.f / (sum + 1e-9f);
#pragma unroll
    for (int t = 0; t < 4; ++t)
      Plds[wv][(r + 8 * g) * 64 + 16 * t + nn] = sv[t] * inv;
  }
  // same-wave LDS RAW: compiler inserts s_wait_dscnt; no barrier needed.

  // ---- O = P * V : 4 dim tiles x 16 K-steps of f32 WMMA ----
  float* Op = O + (b * NQ_ + qbase + wv * 16) * D_ + h * DH_;
#pragma unroll
  for (int dt = 0; dt < 4; ++dt) {
    v8f acc = {0.f, 0.f, 0.f, 0.f, 0.f, 0.f, 0.f, 0.f};
#pragma unroll
    for (int s = 0; s < 16; ++s) {
      v2f ap = *(const v2f*)&Plds[wv][nn * 64 + 4 * s + 2 * g];
      v2f bv = *(const v2f*)&Vlds[(16 * dt + nn) * 64 + 4 * s + 2 * g];
      acc = wmma4(ap, bv, acc);
    }
#pragma unroll
    for (int r = 0; r < 8; ++r)
      Op[(r + 8 * g) * D_ + 16 * dt + nn] = acc[r];
  }
}

extern "C" void kernel_launch(void* const* d_in, const int* in_sizes, int n_in,
                              void* d_out, int out_size, void* d_ws,
                              size_t ws_size, hipStream_t stream) {
  (void)in_sizes; (void)n_in; (void)d_ws; (void)ws_size; (void)out_size;
  const float* q = (const float*)d_in[0];
  const float* k = (const float*)d_in[1];
  const float* v = (const float*)d_in[2];
  // d_in[3] is h (== 8), hardcoded in the kernel specialization.
  float* o = (float*)d_out;
  dim3 grid(NQ_ / QBLK, H_, B_);
  attn_fwd_f32wmma<<<grid, 128, 0, stream>>>(q, k, v, o);
}